// QLSTM_65481071395895
// MI455X (gfx1250) — compile-verified
//
#include <hip/hip_runtime.h>
#include <math.h>

// ---------------------------------------------------------------------------
// QLSTM, analytically simplified:
//   quantum_gate(x, p)[:, i] = prod_{k<=i} cos(p_k) * cos(x_k)
// and only wires 0..3 (the input x, NOT hx) feed the used outputs, so the
// LSTM reduces to an elementwise scan over T per (b, h).
// ---------------------------------------------------------------------------

#define TPB 256

#if __has_builtin(__builtin_amdgcn_global_load_async_to_lds_b128)
#define QL_HAVE_ASYNC_LDS 1
typedef int qv4i __attribute__((ext_vector_type(4)));
typedef __attribute__((address_space(1))) qv4i qg_v4i;  // global
typedef __attribute__((address_space(3))) qv4i ql_v4i;  // LDS
#endif

__device__ __forceinline__ float qsig(float a) {
  return 1.0f / (1.0f + expf(-a));
}

// Phase 1: per (t,b) compute gate values f, u = i*g, o for h = 0..3.
__global__ __launch_bounds__(TPB) void qlstm_gates(
    const float* __restrict__ x,
    const float* __restrict__ pf, const float* __restrict__ pi,
    const float* __restrict__ pg, const float* __restrict__ po,
    float* __restrict__ Fv, float* __restrict__ Uv, float* __restrict__ Ov,
    int nPairs)
{
  int j = blockIdx.x * TPB + threadIdx.x;
  if (j >= nPairs) return;

  // --- load this thread's 4 input angles, staged via CDNA5 async global->LDS
  float4 xv;
#ifdef QL_HAVE_ASYNC_LDS
  __shared__ float tile[TPB * 4];
  __builtin_amdgcn_global_load_async_to_lds_b128(
      (qg_v4i*)(x + 4ll * (long long)j),
      (ql_v4i*)(&tile[threadIdx.x * 4]), 0, 0);
#if __has_builtin(__builtin_amdgcn_s_wait_asynccnt)
  __builtin_amdgcn_s_wait_asynccnt(0);
#else
  asm volatile("s_wait_asynccnt 0" ::: "memory");
#endif
  xv = *(const float4*)(&tile[threadIdx.x * 4]);
#else
  xv = ((const float4*)x)[j];
#endif

  // cumulative products of cos(x_k), k = 0..3
  float c0 = cosf(xv.x);
  float c1 = c0 * cosf(xv.y);
  float c2 = c1 * cosf(xv.z);
  float c3 = c2 * cosf(xv.w);

  // cumulative products of cos(p_k) per gate (wave-uniform scalar work)
  float f0 = cosf(pf[0]); float f1 = f0 * cosf(pf[1]);
  float f2 = f1 * cosf(pf[2]); float f3 = f2 * cosf(pf[3]);
  float i0 = cosf(pi[0]); float i1 = i0 * cosf(pi[1]);
  float i2 = i1 * cosf(pi[2]); float i3 = i2 * cosf(pi[3]);
  float g0 = cosf(pg[0]); float g1 = g0 * cosf(pg[1]);
  float g2 = g1 * cosf(pg[2]); float g3 = g2 * cosf(pg[3]);
  float o0 = cosf(po[0]); float o1 = o0 * cosf(po[1]);
  float o2 = o1 * cosf(po[2]); float o3 = o2 * cosf(po[3]);

  float4 Fo, Uo, Oo;
  Fo.x = qsig(f0 * c0); Fo.y = qsig(f1 * c1);
  Fo.z = qsig(f2 * c2); Fo.w = qsig(f3 * c3);

  Uo.x = qsig(i0 * c0) * tanhf(g0 * c0);
  Uo.y = qsig(i1 * c1) * tanhf(g1 * c1);
  Uo.z = qsig(i2 * c2) * tanhf(g2 * c2);
  Uo.w = qsig(i3 * c3) * tanhf(g3 * c3);

  Oo.x = qsig(o0 * c0); Oo.y = qsig(o1 * c1);
  Oo.z = qsig(o2 * c2); Oo.w = qsig(o3 * c3);

  ((float4*)Fv)[j] = Fo;
  ((float4*)Uv)[j] = Uo;
  ((float4*)Ov)[j] = Oo;
}

// Phase 2: scan over T per element e = b*H + h.
__global__ __launch_bounds__(TPB) void qlstm_scan(
    const float* __restrict__ Fv, const float* __restrict__ Uv,
    const float* __restrict__ Ov,
    float* __restrict__ out, float* __restrict__ hxo, float* __restrict__ cxo,
    int T, int m)
{
  int e = blockIdx.x * TPB + threadIdx.x;
  if (e >= m) return;
  float c = 0.0f, h = 0.0f;
  for (int t = 0; t < T; ++t) {
    long long idx = (long long)t * m + e;
    if (t + 1 < T) {  // prefetch next step's operands (global_prefetch_b8)
      __builtin_prefetch(Fv + idx + m, 0, 0);
      __builtin_prefetch(Uv + idx + m, 0, 0);
      __builtin_prefetch(Ov + idx + m, 0, 0);
    }
    float f = Fv[idx];
    float u = Uv[idx];
    float o = Ov[idx];
    c = fmaf(f, c, u);
    h = o * tanhf(c);
    out[idx] = h;
  }
  hxo[e] = h;
  cxo[e] = c;
}

// Fallback: fused single kernel if workspace is too small.
__global__ __launch_bounds__(TPB) void qlstm_fused(
    const float* __restrict__ x,
    const float* __restrict__ pf, const float* __restrict__ pi,
    const float* __restrict__ pg, const float* __restrict__ po,
    float* __restrict__ out, float* __restrict__ hxo, float* __restrict__ cxo,
    int T, int B)
{
  int e = blockIdx.x * TPB + threadIdx.x;
  int m = B * 4;
  if (e >= m) return;
  int b = e >> 2, h = e & 3;

  float cpf = 1.f, cpi = 1.f, cpg = 1.f, cpo = 1.f;
  for (int k = 0; k <= h; ++k) {
    cpf *= cosf(pf[k]); cpi *= cosf(pi[k]);
    cpg *= cosf(pg[k]); cpo *= cosf(po[k]);
  }

  float c = 0.f, hh = 0.f;
  for (int t = 0; t < T; ++t) {
    long long base = ((long long)t * B + b) * 4;
    float a = 1.f;
    for (int k = 0; k <= h; ++k) a *= cosf(x[base + k]);
    float f = qsig(cpf * a);
    float u = qsig(cpi * a) * tanhf(cpg * a);
    float o = qsig(cpo * a);
    c = fmaf(f, c, u);
    hh = o * tanhf(c);
    out[base + h] = hh;
  }
  hxo[e] = hh;
  cxo[e] = c;
}

extern "C" void kernel_launch(void* const* d_in, const int* in_sizes, int n_in,
                              void* d_out, int out_size, void* d_ws, size_t ws_size,
                              hipStream_t stream) {
  const float* x  = (const float*)d_in[0];
  const float* pf = (const float*)d_in[1];
  const float* pi = (const float*)d_in[2];
  const float* pg = (const float*)d_in[3];
  const float* po = (const float*)d_in[4];
  float* out = (float*)d_out;

  const int D = 4, H = 4;
  long long inN = in_sizes[0];                       // T*B*D
  long long BH2 = (long long)out_size - inN;         // 2*B*H
  int B = (int)(BH2 / (2 * H));                      // 4096
  int T = (int)(inN / ((long long)B * D));           // 128
  int nPairs = T * B;
  int m = B * H;

  float* hxp = out + (long long)nPairs * H;
  float* cxp = hxp + m;

  size_t need = (size_t)nPairs * H * sizeof(float) * 3;
  if (ws_size >= need) {
    float* Fv = (float*)d_ws;
    float* Uv = Fv + (size_t)nPairs * H;
    float* Ov = Uv + (size_t)nPairs * H;

    hipLaunchKernelGGL(qlstm_gates, dim3((nPairs + TPB - 1) / TPB), dim3(TPB),
                       0, stream, x, pf, pi, pg, po, Fv, Uv, Ov, nPairs);
    hipLaunchKernelGGL(qlstm_scan, dim3((m + TPB - 1) / TPB), dim3(TPB),
                       0, stream, Fv, Uv, Ov, out, hxp, cxp, T, m);
  } else {
    hipLaunchKernelGGL(qlstm_fused, dim3((m + TPB - 1) / TPB), dim3(TPB),
                       0, stream, x, pf, pi, pg, po, out, hxp, cxp, T, B);
  }
}